// SAGE_26912265076914
// MI455X (gfx1250) — compile-verified
//
#include <hip/hip_runtime.h>

typedef float v2f __attribute__((ext_vector_type(2)));
typedef float v8f __attribute__((ext_vector_type(8)));

#define WMMA_F32(a, b, c) \
  __builtin_amdgcn_wmma_f32_16x16x4_f32(false, (a), false, (b), (short)0, (c), false, false)

// ---------------------------------------------------------------------------
// GEMM, K=64, two 64x32 weight matrices in one pass.
// One wave32 per 16-row tile of X. Output widths 32.
//   Hself  = X @ Wself + bias
//   Hneigh = X @ Wneigh
// ---------------------------------------------------------------------------
__global__ void __launch_bounds__(256) sage_gemm_k64(
    const float* __restrict__ X,       // [N,64]
    const float* __restrict__ Wself,   // [64,32]
    const float* __restrict__ Wneigh,  // [64,32]
    const float* __restrict__ bias,    // [32]
    float* __restrict__ Hself,         // [N,32]
    float* __restrict__ Hneigh,        // [N,32]
    int ntiles)
{
  int wave = (int)((blockIdx.x * blockDim.x + threadIdx.x) >> 5);
  if (wave >= ntiles) return;                 // wave-uniform: EXEC stays all-1s
  int lane = threadIdx.x & 31;
  int lo   = lane & 15;
  bool hi  = lane >= 16;
  int row0 = wave * 16;
  int arow = row0 + lo;

  v8f s0 = {}; v8f s1 = {}; v8f n0 = {}; v8f n1 = {};
  #pragma unroll
  for (int k = 0; k < 64; k += 4) {
    int ka = k + (hi ? 2 : 0);
    v2f a;
    a.x = X[arow * 64 + ka];
    a.y = X[arow * 64 + ka + 1];
    const float* ws = Wself  + ka * 32;   // row K=ka of W, next row at +32
    const float* wn = Wneigh + ka * 32;
    v2f bs0, bs1, bn0, bn1;
    bs0.x = ws[lo];      bs0.y = ws[32 + lo];
    bs1.x = ws[16 + lo]; bs1.y = ws[48 + lo];
    bn0.x = wn[lo];      bn0.y = wn[32 + lo];
    bn1.x = wn[16 + lo]; bn1.y = wn[48 + lo];
    s0 = WMMA_F32(a, bs0, s0);
    s1 = WMMA_F32(a, bs1, s1);
    n0 = WMMA_F32(a, bn0, n0);
    n1 = WMMA_F32(a, bn1, n1);
  }
  int rbase = row0 + (hi ? 8 : 0);
  float b0 = bias[lo], b1v = bias[16 + lo];
  #pragma unroll
  for (int i = 0; i < 8; ++i) {
    int r = rbase + i;
    Hself[r * 32 + lo]       = s0[i] + b0;
    Hself[r * 32 + 16 + lo]  = s1[i] + b1v;
    Hneigh[r * 32 + lo]      = n0[i];
    Hneigh[r * 32 + 16 + lo] = n1[i];
  }
}

// ---------------------------------------------------------------------------
// GEMM, K=32, two 32x16 weight matrices in one pass. Output widths 16.
// ---------------------------------------------------------------------------
__global__ void __launch_bounds__(256) sage_gemm_k32(
    const float* __restrict__ X,       // [N,32]
    const float* __restrict__ Wself,   // [32,16]
    const float* __restrict__ Wneigh,  // [32,16]
    const float* __restrict__ bias,    // [16]
    float* __restrict__ Oself,         // [N,16]
    float* __restrict__ Oneigh,        // [N,16]
    int ntiles)
{
  int wave = (int)((blockIdx.x * blockDim.x + threadIdx.x) >> 5);
  if (wave >= ntiles) return;
  int lane = threadIdx.x & 31;
  int lo   = lane & 15;
  bool hi  = lane >= 16;
  int row0 = wave * 16;
  int arow = row0 + lo;

  v8f s0 = {}; v8f n0 = {};
  #pragma unroll
  for (int k = 0; k < 32; k += 4) {
    int ka = k + (hi ? 2 : 0);
    v2f a;
    a.x = X[arow * 32 + ka];
    a.y = X[arow * 32 + ka + 1];
    const float* ws = Wself  + ka * 16;
    const float* wn = Wneigh + ka * 16;
    v2f bs, bn;
    bs.x = ws[lo]; bs.y = ws[16 + lo];
    bn.x = wn[lo]; bn.y = wn[16 + lo];
    s0 = WMMA_F32(a, bs, s0);
    n0 = WMMA_F32(a, bn, n0);
  }
  int rbase = row0 + (hi ? 8 : 0);
  float b0 = bias[lo];
  #pragma unroll
  for (int i = 0; i < 8; ++i) {
    int r = rbase + i;
    Oself[r * 16 + lo]  = s0[i] + b0;
    Oneigh[r * 16 + lo] = n0[i];
  }
}

// ---------------------------------------------------------------------------
// In-degree count (float, for mean normalization)
// ---------------------------------------------------------------------------
__global__ void __launch_bounds__(256) sage_deg(
    const int* __restrict__ dst, float* __restrict__ deg, int E)
{
  int i = blockIdx.x * blockDim.x + threadIdx.x;
  if (i >= E) return;
  __builtin_prefetch(dst + i + 8192, 0, 1);
  __hip_atomic_fetch_add(&deg[dst[i]], 1.0f,
                         __ATOMIC_RELAXED, __HIP_MEMORY_SCOPE_AGENT);
}

// ---------------------------------------------------------------------------
// Edge scatter: agg[dst][f] += F[src][f], width = 2^LOGW floats per node.
// (edge, feature) pair per thread -> coalesced 2^LOGW-float segments, atomics
// land in L2 (node tables fit in the 192MB L2).
// ---------------------------------------------------------------------------
template <int LOGW>
__global__ void __launch_bounds__(256) sage_scatter(
    const int* __restrict__ src, const int* __restrict__ dst,
    const float* __restrict__ F, float* __restrict__ agg, int E)
{
  unsigned i = blockIdx.x * blockDim.x + threadIdx.x;
  unsigned e = i >> LOGW;
  unsigned f = i & ((1u << LOGW) - 1u);
  if (e >= (unsigned)E) return;
  if (f == 0) {
    __builtin_prefetch(src + e + 4096, 0, 1);
    __builtin_prefetch(dst + e + 4096, 0, 1);
  }
  int s = src[e], d = dst[e];
  float v = F[((size_t)s << LOGW) + f];
  __hip_atomic_fetch_add(&agg[((size_t)d << LOGW) + f], v,
                         __ATOMIC_RELAXED, __HIP_MEMORY_SCOPE_AGENT);
}

// ---------------------------------------------------------------------------
// H[n][f] += agg[n][f] / max(deg[n], 1)
// ---------------------------------------------------------------------------
template <int LOGW>
__global__ void __launch_bounds__(256) sage_finalize(
    float* __restrict__ H, const float* __restrict__ agg,
    const float* __restrict__ deg, int total)
{
  int i = blockIdx.x * blockDim.x + threadIdx.x;
  if (i >= total) return;
  int node = i >> LOGW;
  H[i] += agg[i] / fmaxf(deg[node], 1.0f);
}

// ---------------------------------------------------------------------------
extern "C" void kernel_launch(void* const* d_in, const int* in_sizes, int n_in,
                              void* d_out, int out_size, void* d_ws, size_t ws_size,
                              hipStream_t stream) {
  const float* x   = (const float*)d_in[0];
  const int*   src = (const int*)  d_in[1];
  const int*   dst = (const int*)  d_in[2];
  const float* Ws1 = (const float*)d_in[3];
  const float* Wn1 = (const float*)d_in[4];
  const float* b1  = (const float*)d_in[5];
  const float* Ws2 = (const float*)d_in[6];
  const float* Wn2 = (const float*)d_in[7];
  const float* b2  = (const float*)d_in[8];
  float* out = (float*)d_out;

  const int N = in_sizes[0] / 64;   // 100000 (divisible by 16)
  const int E = in_sizes[1];        // 1600000
  const int ntiles = N / 16;

  // workspace layout (floats)
  float* ws   = (float*)d_ws;
  float* deg  = ws;                      // N        (pad to 102400)
  float* xw1  = deg  + 102400;           // N*32
  float* agg1 = xw1  + (size_t)N * 32;   // N*32
  float* h    = agg1 + (size_t)N * 32;   // N*32
  float* hw2  = h    + (size_t)N * 32;   // N*16
  float* agg2 = hw2  + (size_t)N * 16;   // N*16

  hipMemsetAsync(deg,  0, (size_t)N * sizeof(float), stream);
  hipMemsetAsync(agg1, 0, (size_t)N * 32 * sizeof(float), stream);
  hipMemsetAsync(agg2, 0, (size_t)N * 16 * sizeof(float), stream);

  const int TB = 256;

  // degrees
  sage_deg<<<(E + TB - 1) / TB, TB, 0, stream>>>(dst, deg, E);

  // layer 1 dense: h = x@Wself1 + b1 ; xw1 = x@Wneigh1   (WMMA f32)
  {
    int threads = ntiles * 32;
    sage_gemm_k64<<<(threads + TB - 1) / TB, TB, 0, stream>>>(
        x, Ws1, Wn1, b1, h, xw1, ntiles);
  }

  // layer 1 edge aggregation on the 32-wide projected features
  {
    long long threads = (long long)E * 32;
    sage_scatter<5><<<(unsigned)((threads + TB - 1) / TB), TB, 0, stream>>>(
        src, dst, xw1, agg1, E);
  }
  sage_finalize<5><<<(N * 32 + TB - 1) / TB, TB, 0, stream>>>(h, agg1, deg, N * 32);

  // layer 2 dense: out = h@Wself2 + b2 ; hw2 = h@Wneigh2  (WMMA f32)
  {
    int threads = ntiles * 32;
    sage_gemm_k32<<<(threads + TB - 1) / TB, TB, 0, stream>>>(
        h, Ws2, Wn2, b2, out, hw2, ntiles);
  }

  // layer 2 edge aggregation on the 16-wide projected features
  {
    long long threads = (long long)E * 16;
    sage_scatter<4><<<(unsigned)((threads + TB - 1) / TB), TB, 0, stream>>>(
        src, dst, hw2, agg2, E);
  }
  sage_finalize<4><<<(N * 16 + TB - 1) / TB, TB, 0, stream>>>(out, agg2, deg, N * 16);
}